// FEModel_82008105550499
// MI455X (gfx1250) — compile-verified
//
#include <hip/hip_runtime.h>
#include <hip/hip_bf16.h>
#include <cmath>

typedef __attribute__((ext_vector_type(2))) float v2f;
typedef __attribute__((ext_vector_type(8))) float v8f;

#define SLOPE 0.2f

// ---------------- utility kernels ----------------

__global__ void fill_kernel(float* __restrict__ p, float v, int n) {
    int i = blockIdx.x * blockDim.x + threadIdx.x;
    if (i < n) p[i] = v;
}

// deg[d] += 1 ; lattr[d] += edge_attr[e]
__global__ void deg_sum_kernel(const int* __restrict__ dst, const float* __restrict__ eattr,
                               float* __restrict__ deg, float* __restrict__ lattr, int E) {
    int e = blockIdx.x * blockDim.x + threadIdx.x;
    if (e >= E) return;
    int d = dst[e];
    atomicAdd(deg + d, 1.0f);
    atomicAdd(lattr + d, eattr[e]);
}

__global__ void loop_div_kernel(float* __restrict__ lattr, const float* __restrict__ deg, int n) {
    int i = blockIdx.x * blockDim.x + threadIdx.x;
    if (i >= n) return;
    lattr[i] = lattr[i] / fmaxf(deg[i], 1.0f);
}

// x[i,0:16] = emb[ids[i]] ; x[i,16] = feat[i]   (row stride 17)
__global__ void embed_concat_kernel(const int* __restrict__ ids, const float* __restrict__ feat,
                                    const float* __restrict__ emb, float* __restrict__ X, int n) {
    int i = blockIdx.x * blockDim.x + threadIdx.x;
    if (i >= n) return;
    const float4* ev = (const float4*)(emb + (size_t)ids[i] * 16);
    float4 e0 = ev[0], e1 = ev[1], e2 = ev[2], e3 = ev[3];
    float* xr = X + (size_t)i * 17;
    xr[0]=e0.x; xr[1]=e0.y; xr[2]=e0.z;  xr[3]=e0.w;
    xr[4]=e1.x; xr[5]=e1.y; xr[6]=e1.z;  xr[7]=e1.w;
    xr[8]=e2.x; xr[9]=e2.y; xr[10]=e2.z; xr[11]=e2.w;
    xr[12]=e3.x; xr[13]=e3.y; xr[14]=e3.z; xr[15]=e3.w;
    xr[16] = feat[i];
}

// out[0] = dot(a[0:32], b[0:32]) ; launched with one wave32
__global__ void dot32_kernel(const float* __restrict__ a, const float* __restrict__ b,
                             float* __restrict__ out) {
    int l = threadIdx.x;
    float p = a[l] * b[l];
    #pragma unroll
    for (int off = 16; off > 0; off >>= 1) p += __shfl_xor(p, off, 32);
    if (l == 0) out[0] = p;
}

// ---------------- WMMA GEMM:  H[N,32] = X[N,K] @ W[K,32]  ----------------
// One wave computes a 16-row tile: two f32 16x16 accumulators (cols 0-15, 16-31).
// A fragment (16x4 f32): lane m(0-15)+half hi: v0 = X[row0+m][k0+2*hi], v1 = X[..][k0+2*hi+1]
// B fragment (4x16 f32): v0 = W[k0+2*hi][n0+m],  v1 = W[k0+2*hi+1][n0+m]
// C/D: vgpr r <-> row row0 + r + 8*hi, col n0 + m.
__global__ void gemm_xw_kernel(const float* __restrict__ X, const float* __restrict__ W,
                               float* __restrict__ H, int ntiles, int K) {
    const int lane = threadIdx.x & 31;
    const int wave = blockIdx.x * (blockDim.x >> 5) + (threadIdx.x >> 5);
    if (wave >= ntiles) return;           // wave-uniform; EXEC stays all-ones for WMMA
    const int row0 = wave << 4;
    const int m  = lane & 15;
    const int hi = lane >> 4;

    v8f c0 = {}; v8f c1 = {};
    const int K4 = K & ~3;
    for (int k0 = 0; k0 < K4; k0 += 4) {
        const int ka = k0 + 2 * hi;
        v2f a, b0, b1;
        const float* xr = X + (size_t)(row0 + m) * K + ka;
        a.x = xr[0]; a.y = xr[1];
        const float* w0 = W + (size_t)ka * 32;
        b0.x = w0[m];      b0.y = w0[32 + m];
        b1.x = w0[16 + m]; b1.y = w0[48 + m];
        c0 = __builtin_amdgcn_wmma_f32_16x16x4_f32(false, a, false, b0, (short)0, c0, false, false);
        c1 = __builtin_amdgcn_wmma_f32_16x16x4_f32(false, a, false, b1, (short)0, c1, false, false);
    }
    for (int k = K4; k < K; ++k) {        // rank-1 tail (K=17: the scalar-feature column)
        const float w0 = W[(size_t)k * 32 + m];
        const float w1 = W[(size_t)k * 32 + 16 + m];
        #pragma unroll
        for (int r = 0; r < 8; ++r) {
            const float xv = X[(size_t)(row0 + r + 8 * hi) * K + k];
            c0[r] += xv * w0;
            c1[r] += xv * w1;
        }
    }
    #pragma unroll
    for (int r = 0; r < 8; ++r) {
        float* hr = H + (size_t)(row0 + r + 8 * hi) * 32;
        hr[m]      = c0[r];
        hr[16 + m] = c1[r];
    }
}

// scalar fallback for rows not covered by full 16-row tiles (0 rows for N=100000)
__global__ void gemm_tail_kernel(const float* __restrict__ X, const float* __restrict__ W,
                                 float* __restrict__ H, int row_start, int nrows, int K) {
    int t = blockIdx.x * blockDim.x + threadIdx.x;
    if (t >= (nrows - row_start) * 32) return;
    int row = row_start + (t >> 5), c = t & 31;
    float s = 0.f;
    for (int k = 0; k < K; ++k) s += X[(size_t)row * K + k] * W[(size_t)k * 32 + c];
    H[(size_t)row * 32 + c] = s;
}

// ---------------- per-node attention logits ----------------
__global__ void alpha_node_kernel(const float* __restrict__ H, const float* __restrict__ a_s,
                                  const float* __restrict__ a_d, float* __restrict__ asrc,
                                  float* __restrict__ adst, int n) {
    int i = blockIdx.x * blockDim.x + threadIdx.x;
    if (i >= n) return;
    const float4* hv = (const float4*)(H + (size_t)i * 32);
    float ss = 0.f, sd = 0.f;
    #pragma unroll
    for (int q = 0; q < 8; ++q) {
        float4 h4 = hv[q];
        ss += h4.x * a_s[4*q] + h4.y * a_s[4*q+1] + h4.z * a_s[4*q+2] + h4.w * a_s[4*q+3];
        sd += h4.x * a_d[4*q] + h4.y * a_d[4*q+1] + h4.z * a_d[4*q+2] + h4.w * a_d[4*q+3];
    }
    asrc[i] = ss; adst[i] = sd;
}

// ---------------- edge passes ----------------
__device__ __forceinline__ void atomicMaxF(float* addr, float val) {
    if (val >= 0.f) atomicMax((int*)addr, __float_as_int(val));
    else            atomicMin((unsigned int*)addr, __float_as_uint(val));
}

// pass 1: alpha = leaky_relu(asrc[s]+adst[d]+ce*ea) ; segment max into mbuf[d]
__global__ void edge_alpha_max_kernel(const int* __restrict__ src, const int* __restrict__ dst,
                                      const float* __restrict__ eattr, const float* __restrict__ lattr,
                                      const float* __restrict__ asrc, const float* __restrict__ adst,
                                      const float* __restrict__ ce, float* __restrict__ alpha,
                                      float* __restrict__ mbuf, int E, int E2) {
    int e = blockIdx.x * blockDim.x + threadIdx.x;
    if (e >= E2) return;
    int s, d; float ea;
    if (e < E) { s = src[e]; d = dst[e]; ea = eattr[e]; }
    else       { s = d = e - E;          ea = lattr[e - E]; }
    float a = asrc[s] + adst[d] + ce[0] * ea;
    a = a > 0.f ? a : SLOPE * a;
    alpha[e] = a;
    atomicMaxF(mbuf + d, a);
}

// pass 2: ex = exp(alpha - m[d]) ; alpha <- ex ; denom[d] += ex
__global__ void edge_exp_sum_kernel(const int* __restrict__ dst, float* __restrict__ alpha,
                                    const float* __restrict__ mbuf, float* __restrict__ denom,
                                    int E, int E2) {
    int e = blockIdx.x * blockDim.x + threadIdx.x;
    if (e >= E2) return;
    int d = (e < E) ? dst[e] : (e - E);
    float ex = __expf(alpha[e] - mbuf[d]);
    alpha[e] = ex;
    atomicAdd(denom + d, ex);
}

// pass 3: one wave per edge; agg[d, lane] += (ex/denom[d]) * H[s, lane]
__global__ void edge_aggregate_kernel(const int* __restrict__ src, const int* __restrict__ dst,
                                      const float* __restrict__ ex, const float* __restrict__ denom,
                                      const float* __restrict__ H, float* __restrict__ agg,
                                      int E, int E2) {
    int t = blockIdx.x * blockDim.x + threadIdx.x;
    int e = t >> 5, c = t & 31;
    if (e >= E2) return;
    int s, d;
    if (e < E) { s = src[e]; d = dst[e]; }
    else       { s = d = e - E; }
    float coef = ex[e] / (denom[d] + 1e-16f);
    atomicAdd(agg + (size_t)d * 32 + c, coef * H[(size_t)s * 32 + c]);
}

__global__ void bias_relu_kernel(float* __restrict__ agg, const float* __restrict__ b, int n32) {
    int t = blockIdx.x * blockDim.x + threadIdx.x;
    if (t >= n32) return;
    agg[t] = fmaxf(agg[t] + b[t & 31], 0.f);
}

// out[i] = sum_c (agg[i,c] + b2[c]) * lin_w[c] + lin_b
__global__ void final_linear_kernel(const float* __restrict__ agg, const float* __restrict__ b,
                                    const float* __restrict__ lw, const float* __restrict__ lb,
                                    float* __restrict__ out, int n) {
    int i = blockIdx.x * blockDim.x + threadIdx.x;
    if (i >= n) return;
    const float4* av = (const float4*)(agg + (size_t)i * 32);
    float s = lb[0];
    #pragma unroll
    for (int q = 0; q < 8; ++q) {
        float4 a4 = av[q];
        s += (a4.x + b[4*q  ]) * lw[4*q  ] + (a4.y + b[4*q+1]) * lw[4*q+1]
           + (a4.z + b[4*q+2]) * lw[4*q+2] + (a4.w + b[4*q+3]) * lw[4*q+3];
    }
    out[i] = s;
}

// ---------------- launcher ----------------
static inline int cdiv(long long a, long long b) { return (int)((a + b - 1) / b); }

extern "C" void kernel_launch(void* const* d_in, const int* in_sizes, int n_in,
                              void* d_out, int out_size, void* d_ws, size_t ws_size,
                              hipStream_t stream) {
    const int N  = in_sizes[0];
    const int E  = in_sizes[2];
    const int E2 = E + N;

    const int*   x_ids  = (const int*)d_in[0];
    const float* x_feat = (const float*)d_in[1];
    const int*   src    = (const int*)d_in[2];
    const int*   dst    = (const int*)d_in[3];
    const float* eattr  = (const float*)d_in[4];
    const float* emb    = (const float*)d_in[5];
    const float* W1     = (const float*)d_in[6];
    const float* a_s1   = (const float*)d_in[7];
    const float* a_d1   = (const float*)d_in[8];
    const float* We1    = (const float*)d_in[9];
    const float* a_e1   = (const float*)d_in[10];
    const float* b1     = (const float*)d_in[11];
    const float* W2     = (const float*)d_in[12];
    const float* a_s2   = (const float*)d_in[13];
    const float* a_d2   = (const float*)d_in[14];
    const float* We2    = (const float*)d_in[15];
    const float* a_e2   = (const float*)d_in[16];
    const float* b2     = (const float*)d_in[17];
    const float* lin_w  = (const float*)d_in[18];
    const float* lin_b  = (const float*)d_in[19];
    float* out = (float*)d_out;

    // workspace carve-up (all offsets 16B aligned for the sizes involved)
    float* ws    = (float*)d_ws;
    float* X     = ws; ws += (size_t)N * 17;
    float* H     = ws; ws += (size_t)N * 32;   // reused by both layers
    float* AGG1  = ws; ws += (size_t)N * 32;   // layer-1 out -> relu -> layer-2 in
    float* AGG2  = ws; ws += (size_t)N * 32;
    float* ALPHA = ws; ws += (size_t)E2;
    float* ASRC  = ws; ws += N;
    float* ADST  = ws; ws += N;
    float* MBUF  = ws; ws += N;
    float* DEN   = ws; ws += N;
    float* DEG   = ws; ws += N;
    float* LATTR = ws; ws += N;
    float* CE1   = ws; ws += 4;
    float* CE2   = ws; ws += 4;

    const int B = 256;
    const int ntiles = N >> 4;
    const int tail   = N & 15;

    // ---- self-loop edge attr (fill_value='mean') ----
    fill_kernel<<<cdiv(N, B), B, 0, stream>>>(DEG, 0.f, N);
    fill_kernel<<<cdiv(N, B), B, 0, stream>>>(LATTR, 0.f, N);
    deg_sum_kernel<<<cdiv(E, B), B, 0, stream>>>(dst, eattr, DEG, LATTR, E);
    loop_div_kernel<<<cdiv(N, B), B, 0, stream>>>(LATTR, DEG, N);

    // ---- node features + edge-logit constants ----
    embed_concat_kernel<<<cdiv(N, B), B, 0, stream>>>(x_ids, x_feat, emb, X, N);
    dot32_kernel<<<1, 32, 0, stream>>>(We1, a_e1, CE1);
    dot32_kernel<<<1, 32, 0, stream>>>(We2, a_e2, CE2);

    for (int layer = 0; layer < 2; ++layer) {
        const float* Xin = (layer == 0) ? X : AGG1;
        const int    K   = (layer == 0) ? 17 : 32;
        const float* W   = (layer == 0) ? W1 : W2;
        const float* as_ = (layer == 0) ? a_s1 : a_s2;
        const float* ad_ = (layer == 0) ? a_d1 : a_d2;
        const float* CE  = (layer == 0) ? CE1 : CE2;
        float*       AGG = (layer == 0) ? AGG1 : AGG2;

        gemm_xw_kernel<<<cdiv(ntiles, B / 32), B, 0, stream>>>(Xin, W, H, ntiles, K);
        if (tail)
            gemm_tail_kernel<<<cdiv((long long)tail * 32, B), B, 0, stream>>>(Xin, W, H, ntiles * 16, N, K);

        alpha_node_kernel<<<cdiv(N, B), B, 0, stream>>>(H, as_, ad_, ASRC, ADST, N);

        fill_kernel<<<cdiv(N, B), B, 0, stream>>>(MBUF, -INFINITY, N);
        fill_kernel<<<cdiv(N, B), B, 0, stream>>>(DEN, 0.f, N);
        fill_kernel<<<cdiv((long long)N * 32, B), B, 0, stream>>>(AGG, 0.f, N * 32);

        edge_alpha_max_kernel<<<cdiv(E2, B), B, 0, stream>>>(src, dst, eattr, LATTR, ASRC, ADST,
                                                             CE, ALPHA, MBUF, E, E2);
        edge_exp_sum_kernel<<<cdiv(E2, B), B, 0, stream>>>(dst, ALPHA, MBUF, DEN, E, E2);
        edge_aggregate_kernel<<<cdiv((long long)E2 * 32, B), B, 0, stream>>>(src, dst, ALPHA, DEN,
                                                                             H, AGG, E, E2);
        if (layer == 0)
            bias_relu_kernel<<<cdiv((long long)N * 32, B), B, 0, stream>>>(AGG1, b1, N * 32);
    }

    final_linear_kernel<<<cdiv(N, B), B, 0, stream>>>(AGG2, b2, lin_w, lin_b, out, N);
    (void)n_in; (void)out_size; (void)ws_size;
}